// GraphNet_39848706572460
// MI455X (gfx1250) — compile-verified
//
#include <hip/hip_runtime.h>
#include <cstring>
#include <cstddef>

// ---------------------------------------------------------------------------
// GraphNet forward for MI455X (gfx1250): 3 x (TransformerConv -> LN/FFN ->
// SAGPool -> readout) + BN-MLP head.  Dense GEMMs use V_WMMA_F32_16X16X4_F32
// (fp32 matrix pipe, exact numerics; GEMMs are not the roofline bottleneck --
// the L2-resident edge scatters are).
// ---------------------------------------------------------------------------

#define Bg   64
#define N0g  400
#define Dg   128
#define Hg   4
#define Cg   32
#define Eg   409600     // total edges (fixed across stages; mask shrinks)
#define HIDg 64

typedef __attribute__((ext_vector_type(2))) float v2f;
typedef __attribute__((ext_vector_type(8))) float v8f;

// ---- float <-> order-preserving uint (for atomic max on floats) -----------
__device__ __forceinline__ unsigned fenc(float f) {
  unsigned u = __float_as_uint(f);
  return (u & 0x80000000u) ? ~u : (u | 0x80000000u);
}
__device__ __forceinline__ float fdec(unsigned u) {
  unsigned v = (u & 0x80000000u) ? (u & 0x7fffffffu) : ~u;
  return __uint_as_float(v);
}
static inline unsigned host_fenc(float f) {
  unsigned u; memcpy(&u, &f, 4);
  return (u & 0x80000000u) ? ~u : (u | 0x80000000u);
}

// ---- init helpers ---------------------------------------------------------
__global__ void k_init_f32(float* p, int n, float v) {
  int i = blockIdx.x * blockDim.x + threadIdx.x;
  if (i < n) p[i] = v;
}
__global__ void k_init_u32(unsigned* p, int n, unsigned v) {
  int i = blockIdx.x * blockDim.x + threadIdx.x;
  if (i < n) p[i] = v;
}
__global__ void k_edge_init(const int* es, const int* ed, int* s, int* d, int* m, int E) {
  int e = blockIdx.x * blockDim.x + threadIdx.x;
  if (e >= E) return;
  s[e] = es[e]; d[e] = ed[e]; m[e] = 1;
}

// ---- x = x + lap @ W_pos + b_pos ------------------------------------------
__global__ void k_pos_encode(const float* __restrict__ x, const float* __restrict__ lap,
                             const float* __restrict__ Wp, const float* __restrict__ bp,
                             float* __restrict__ out, int N) {
  int i = blockIdx.x * blockDim.x + threadIdx.x;
  if (i >= N * Dg) return;
  int node = i >> 7, d = i & 127;
  float acc = bp[d];
  const float* l = lap + node * 16;
#pragma unroll
  for (int p = 0; p < 16; p++) acc += l[p] * Wp[p * Dg + d];
  out[i] = x[i] + acc;
}

// ---- WMMA GEMM: Y[N,Dout] = X[N,Din] @ W[Din,Dout] + bias, optional ReLU --
// One wave per 16x16 output tile; K stepped by 4 (V_WMMA_F32_16X16X4_F32).
// Operand layout per CDNA5 ISA 7.12.2:
//   A 16x4: lanes 0-15 hold K={0,1}, lanes 16-31 hold K={2,3} (v0/v1)
//   B 4x16: lanes 0-15 hold rows K={0,1}, lanes 16-31 rows K={2,3}
//   C/D:    vgpr r -> row r (lanes 0-15) / row r+8 (lanes 16-31), col = lane%16
__global__ void k_gemm_wmma(const float* __restrict__ X, const float* __restrict__ W,
                            const float* __restrict__ bias, float* __restrict__ Y,
                            int N, int Din, int Dout, int relu) {
  int wave = (blockIdx.x * blockDim.x + threadIdx.x) >> 5;
  int lane = threadIdx.x & 31;
  int tilesN = Dout >> 4;
  int tilesM = N >> 4;
  if (wave >= tilesM * tilesN) return;
  int tm = wave / tilesN, tn = wave % tilesN;
  int row0 = tm << 4, col0 = tn << 4;
  int half = lane >> 4, lp = lane & 15;

  v8f c = {};
  const float* xr = X + (size_t)(row0 + lp) * Din;
  const float* wc = W + col0 + lp;
  for (int k0 = 0; k0 < Din; k0 += 4) {
    int ka = k0 + 2 * half;
    v2f a, b;
    a.x = xr[ka];
    a.y = xr[ka + 1];
    b.x = wc[(size_t)ka * Dout];
    b.y = wc[(size_t)(ka + 1) * Dout];
    c = __builtin_amdgcn_wmma_f32_16x16x4_f32(false, a, false, b, (short)0, c,
                                              false, false);
  }
  float bb = bias ? bias[col0 + lp] : 0.0f;
#pragma unroll
  for (int r = 0; r < 8; r++) {
    float o = c[r] + bb;
    if (relu) o = fmaxf(o, 0.0f);
    Y[(size_t)(row0 + r + 8 * half) * Dout + col0 + lp] = o;
  }
}

// ---- fused residual LayerNorm: Y = LN(A + B) * g + beta (rows of 128) -----
__global__ void k_ln_res(const float* __restrict__ A, const float* __restrict__ Bv,
                         const float* __restrict__ g, const float* __restrict__ beta,
                         float* __restrict__ Y, int N) {
  int wave = (blockIdx.x * blockDim.x + threadIdx.x) >> 5;
  int lane = threadIdx.x & 31;
  if (wave >= N) return;
  const float* a = A + (size_t)wave * Dg;
  const float* b = Bv + (size_t)wave * Dg;
  float t[4], s = 0.0f;
#pragma unroll
  for (int j = 0; j < 4; j++) { t[j] = a[lane + 32 * j] + b[lane + 32 * j]; s += t[j]; }
#pragma unroll
  for (int o = 16; o >= 1; o >>= 1) s += __shfl_xor(s, o, 32);
  float mu = s * (1.0f / 128.0f);
  float vs = 0.0f;
#pragma unroll
  for (int j = 0; j < 4; j++) { float d = t[j] - mu; vs += d * d; }
#pragma unroll
  for (int o = 16; o >= 1; o >>= 1) vs += __shfl_xor(vs, o, 32);
  float inv = rsqrtf(vs * (1.0f / 128.0f) + 1e-5f);
  float* y = Y + (size_t)wave * Dg;
#pragma unroll
  for (int j = 0; j < 4; j++) {
    int d = lane + 32 * j;
    y[d] = (t[j] - mu) * inv * g[d] + beta[d];
  }
}

// ---- attention pass A: per-(edge,head) logit + atomic max over dst --------
__global__ void k_attn_logits(const float* __restrict__ Q, const float* __restrict__ K,
                              const int* __restrict__ src, const int* __restrict__ dst,
                              const int* __restrict__ mask, float* __restrict__ logits,
                              unsigned* __restrict__ nmax, int E) {
  int t = blockIdx.x * blockDim.x + threadIdx.x;
  if (t >= E * Hg) return;
  int e = t >> 2, h = t & 3;
  int s = src[e], d = dst[e];
  const float* q = Q + (size_t)d * Dg + h * Cg;
  const float* k = K + (size_t)s * Dg + h * Cg;
  float acc = 0.0f;
#pragma unroll
  for (int c = 0; c < Cg; c++) acc += q[c] * k[c];
  float l = mask[e] ? acc * 0.17677669529663687f : -1e9f;  // 1/sqrt(32)
  logits[t] = l;
  atomicMax(nmax + d * Hg + h, fenc(l));
}

// ---- attention pass B: w = exp(l - max)*mask; denom += w (in-place) -------
__global__ void k_attn_weights(const int* __restrict__ dst, const int* __restrict__ mask,
                               float* __restrict__ logw, const unsigned* __restrict__ nmax,
                               float* __restrict__ den, int E) {
  int t = blockIdx.x * blockDim.x + threadIdx.x;
  if (t >= E * Hg) return;
  int e = t >> 2, h = t & 3;
  int d = dst[e];
  float w = mask[e] ? __expf(logw[t] - fdec(nmax[d * Hg + h])) : 0.0f;
  logw[t] = w;
  if (w != 0.0f) atomicAdd(den + d * Hg + h, w);
}

// ---- attention pass C: msg[dst] += v[src] * alpha (per edge,feature) ------
__global__ void k_attn_scatter(const float* __restrict__ V, const int* __restrict__ src,
                               const int* __restrict__ dst, const float* __restrict__ wbuf,
                               const float* __restrict__ den, float* __restrict__ msg, int E) {
  int t = blockIdx.x * blockDim.x + threadIdx.x;
  if (t >= E * Dg) return;
  int e = t >> 7, f = t & 127, h = f >> 5;
  float w = wbuf[e * Hg + h];
  if (w == 0.0f) return;
  int d = dst[e], s = src[e];
  float dn = den[d * Hg + h];
  float alpha = w / (dn > 0.0f ? dn : 1.0f);
  atomicAdd(msg + (size_t)d * Dg + f, V[(size_t)s * Dg + f] * alpha);
}

// ---- SAGPool: agg[dst] += x[src]*mask -------------------------------------
__global__ void k_pool_agg(const float* __restrict__ X, const int* __restrict__ src,
                           const int* __restrict__ dst, const int* __restrict__ mask,
                           float* __restrict__ agg, int E) {
  int t = blockIdx.x * blockDim.x + threadIdx.x;
  if (t >= E * Dg) return;
  int e = t >> 7, f = t & 127;
  if (!mask[e]) return;
  atomicAdd(agg + (size_t)dst[e] * Dg + f, X[(size_t)src[e] * Dg + f]);
}

// ---- SAGPool: score = tanh(agg@Wrel + brel + x@Wroot), wave per node ------
__global__ void k_pool_score(const float* __restrict__ X, const float* __restrict__ agg,
                             const float* __restrict__ Wrel, const float* __restrict__ brel,
                             const float* __restrict__ Wroot, float* __restrict__ score, int N) {
  int wave = (blockIdx.x * blockDim.x + threadIdx.x) >> 5;
  int lane = threadIdx.x & 31;
  if (wave >= N) return;
  float acc = 0.0f;
#pragma unroll
  for (int j = 0; j < 4; j++) {
    int d = lane + 32 * j;
    acc += agg[(size_t)wave * Dg + d] * Wrel[d] + X[(size_t)wave * Dg + d] * Wroot[d];
  }
#pragma unroll
  for (int o = 16; o >= 1; o >>= 1) acc += __shfl_xor(acc, o, 32);
  if (lane == 0) score[wave] = tanhf(acc + brel[0]);
}

// ---- SAGPool: stable rank-by-count top-k (matches jax.lax.top_k order) ----
__global__ void k_pool_rank(const float* __restrict__ score, int* __restrict__ kept,
                            int* __restrict__ newid, int n, int k, int Ntot) {
  int i = blockIdx.x * blockDim.x + threadIdx.x;
  if (i >= Ntot) return;
  int b = i / n, li = i - b * n;
  const float* s = score + b * n;
  float si = s[li];
  int rank = 0;
  for (int j = 0; j < n; j++) {
    float sj = s[j];
    rank += (sj > si) || (sj == si && j < li);
  }
  int kp = rank < k;
  kept[i] = kp;
  newid[i] = kp ? (b * k + rank) : 0;
}

__global__ void k_pool_copy(const float* __restrict__ X, const float* __restrict__ score,
                            const int* __restrict__ kept, const int* __restrict__ newid,
                            float* __restrict__ Xn, int Ntot) {
  int t = blockIdx.x * blockDim.x + threadIdx.x;
  if (t >= Ntot * Dg) return;
  int i = t >> 7, f = t & 127;
  if (!kept[i]) return;
  Xn[(size_t)newid[i] * Dg + f] = X[t] * score[i];
}

__global__ void k_pool_remap(int* __restrict__ src, int* __restrict__ dst, int* __restrict__ mask,
                             const int* __restrict__ kept, const int* __restrict__ newid, int E) {
  int e = blockIdx.x * blockDim.x + threadIdx.x;
  if (e >= E) return;
  int s = src[e], d = dst[e];
  int m = mask[e] && kept[s] && kept[d];
  src[e] = m ? newid[s] : 0;
  dst[e] = m ? newid[d] : 0;
  mask[e] = m;
}

// ---- readout: xf[b, :128] += max over nodes; xf[b,128:] += mean -----------
__global__ void k_readout(const float* __restrict__ X, float* __restrict__ xf, int k) {
  int t = blockIdx.x * blockDim.x + threadIdx.x;
  if (t >= Bg * Dg) return;
  int b = t >> 7, f = t & 127;
  float mx = -1e30f, sm = 0.0f;
  for (int i = 0; i < k; i++) {
    float v = X[((size_t)b * k + i) * Dg + f];
    mx = fmaxf(mx, v);
    sm += v;
  }
  xf[b * 256 + f] += mx;
  xf[b * 256 + 128 + f] += sm / (float)k;
}

// ---- BatchNorm1d (train stats over batch=64) + ReLU, one block per col ----
__global__ void k_bn_relu(float* __restrict__ X, const float* __restrict__ g,
                          const float* __restrict__ b, int cols) {
  int c = blockIdx.x, r = threadIdx.x;  // r in [0,64)
  __shared__ float sh[64];
  float v = X[r * cols + c];
  sh[r] = v;
  __syncthreads();
  for (int o = 32; o >= 1; o >>= 1) { if (r < o) sh[r] += sh[r + o]; __syncthreads(); }
  float mu = sh[0] * (1.0f / 64.0f);
  __syncthreads();
  float dlt = v - mu;
  sh[r] = dlt * dlt;
  __syncthreads();
  for (int o = 32; o >= 1; o >>= 1) { if (r < o) sh[r] += sh[r + o]; __syncthreads(); }
  float var = sh[0] * (1.0f / 64.0f);
  float y = dlt * rsqrtf(var + 1e-5f) * g[c] + b[c];
  X[r * cols + c] = fmaxf(y, 0.0f);
}

__global__ void k_head_final(const float* __restrict__ H2, const float* __restrict__ w,
                             const float* __restrict__ b, float* __restrict__ out) {
  int r = blockIdx.x * blockDim.x + threadIdx.x;
  if (r >= Bg) return;
  float acc = b[0];
#pragma unroll
  for (int j = 0; j < 64; j++) acc += H2[r * 64 + j] * w[j];
  out[r] = acc;
}

// ---------------------------------------------------------------------------
extern "C" void kernel_launch(void* const* d_in, const int* in_sizes, int n_in,
                              void* d_out, int out_size, void* d_ws, size_t ws_size,
                              hipStream_t stream) {
  (void)in_sizes; (void)n_in; (void)out_size; (void)ws_size;
  const float* x        = (const float*)d_in[0];
  const float* lap      = (const float*)d_in[1];
  const int*   e_src    = (const int*)d_in[2];
  const int*   e_dst    = (const int*)d_in[3];
  const float* W_pos    = (const float*)d_in[4];
  const float* b_pos    = (const float*)d_in[5];
  const float* gt_Wq    = (const float*)d_in[6];
  const float* gt_bq    = (const float*)d_in[7];
  const float* gt_Wk    = (const float*)d_in[8];
  const float* gt_bk    = (const float*)d_in[9];
  const float* gt_Wv    = (const float*)d_in[10];
  const float* gt_bv    = (const float*)d_in[11];
  const float* gt_Wsk   = (const float*)d_in[12];
  const float* gt_bsk   = (const float*)d_in[13];
  const float* gt_Wlo   = (const float*)d_in[14];
  const float* gt_blo   = (const float*)d_in[15];
  const float* gt_ln1g  = (const float*)d_in[16];
  const float* gt_ln1b  = (const float*)d_in[17];
  const float* gt_W1    = (const float*)d_in[18];
  const float* gt_b1    = (const float*)d_in[19];
  const float* gt_W2    = (const float*)d_in[20];
  const float* gt_b2    = (const float*)d_in[21];
  const float* gt_ln2g  = (const float*)d_in[22];
  const float* gt_ln2b  = (const float*)d_in[23];
  const float* p_Wrel   = (const float*)d_in[24];
  const float* p_brel   = (const float*)d_in[25];
  const float* p_Wroot  = (const float*)d_in[26];
  const float* hW1      = (const float*)d_in[27];
  const float* hb1      = (const float*)d_in[28];
  const float* hW2      = (const float*)d_in[29];
  const float* hb2      = (const float*)d_in[30];
  const float* hW3      = (const float*)d_in[31];
  const float* hb3      = (const float*)d_in[32];
  const float* bn1g     = (const float*)d_in[33];
  const float* bn1b     = (const float*)d_in[34];
  const float* bn2g     = (const float*)d_in[35];
  const float* bn2b     = (const float*)d_in[36];
  float* out = (float*)d_out;

  const int NMAXN = Bg * N0g;  // 25600
  // workspace carve-up (256B aligned)
  char* wp = (char*)d_ws;
  auto alloc = [&](size_t bytes) -> void* {
    void* p = (void*)wp;
    wp += (bytes + 255) & ~(size_t)255;
    return p;
  };
  float* XA   = (float*)alloc((size_t)NMAXN * Dg * 4);
  float* XB   = (float*)alloc((size_t)NMAXN * Dg * 4);
  float* Qb   = (float*)alloc((size_t)NMAXN * Dg * 4);   // also out1
  float* Kb   = (float*)alloc((size_t)NMAXN * Dg * 4);   // also FFN hidden
  float* Vb   = (float*)alloc((size_t)NMAXN * Dg * 4);   // also out3
  float* MSG  = (float*)alloc((size_t)NMAXN * Dg * 4);   // skip+messages / pool agg
  float* LOGW = (float*)alloc((size_t)Eg * Hg * 4);      // logits -> exp weights
  unsigned* NM = (unsigned*)alloc((size_t)NMAXN * Hg * 4);
  float* DEN  = (float*)alloc((size_t)NMAXN * Hg * 4);
  int* SRC    = (int*)alloc((size_t)Eg * 4);
  int* DST    = (int*)alloc((size_t)Eg * 4);
  int* MASK   = (int*)alloc((size_t)Eg * 4);
  float* SCORE = (float*)alloc((size_t)NMAXN * 4);
  int* KEPT   = (int*)alloc((size_t)NMAXN * 4);
  int* NEWID  = (int*)alloc((size_t)NMAXN * 4);
  float* XF   = (float*)alloc((size_t)Bg * 256 * 4);
  float* H1   = (float*)alloc((size_t)Bg * 128 * 4);
  float* H2   = (float*)alloc((size_t)Bg * 64 * 4);

  auto cdiv = [](long a, long b) { return (int)((a + b - 1) / b); };
  const int T = 256;
  auto gemm = [&](const float* X_, const float* W_, const float* bias_, float* Y_,
                  int N_, int Din_, int Dout_, int relu_) {
    int waves = (N_ >> 4) * (Dout_ >> 4);
    k_gemm_wmma<<<cdiv((long)waves * 32, T), T, 0, stream>>>(X_, W_, bias_, Y_, N_, Din_, Dout_, relu_);
  };

  const int nper[3] = {400, 320, 256};
  const int kout[3] = {320, 256, 205};

  // init
  k_init_f32<<<cdiv(Bg * 256, T), T, 0, stream>>>(XF, Bg * 256, 0.0f);
  k_edge_init<<<cdiv(Eg, T), T, 0, stream>>>(e_src, e_dst, SRC, DST, MASK, Eg);
  k_pos_encode<<<cdiv((long)NMAXN * Dg, T), T, 0, stream>>>(x, lap, W_pos, b_pos, XA, NMAXN);

  float* xc = XA;
  float* xn = XB;
  const unsigned NEG9 = host_fenc(-1e9f);

  for (int i = 0; i < 3; i++) {
    const int Nn = Bg * nper[i];
    const size_t oDD = (size_t)i * Dg * Dg, oD = (size_t)i * Dg;
    const size_t oDH = (size_t)i * Dg * HIDg, oH = (size_t)i * HIDg;

    // projections
    gemm(xc, gt_Wq + oDD, gt_bq + oD, Qb, Nn, Dg, Dg, 0);
    gemm(xc, gt_Wk + oDD, gt_bk + oD, Kb, Nn, Dg, Dg, 0);
    gemm(xc, gt_Wv + oDD, gt_bv + oD, Vb, Nn, Dg, Dg, 0);
    gemm(xc, gt_Wsk + oDD, gt_bsk + oD, MSG, Nn, Dg, Dg, 0);  // skip term seeds msg

    // scatter-softmax attention
    k_init_u32<<<cdiv(Nn * Hg, T), T, 0, stream>>>(NM, Nn * Hg, NEG9);
    k_init_f32<<<cdiv(Nn * Hg, T), T, 0, stream>>>(DEN, Nn * Hg, 0.0f);
    k_attn_logits<<<cdiv((long)Eg * Hg, T), T, 0, stream>>>(Qb, Kb, SRC, DST, MASK, LOGW, NM, Eg);
    k_attn_weights<<<cdiv((long)Eg * Hg, T), T, 0, stream>>>(DST, MASK, LOGW, NM, DEN, Eg);
    k_attn_scatter<<<cdiv((long)Eg * Dg, T), T, 0, stream>>>(Vb, SRC, DST, LOGW, DEN, MSG, Eg);

    // out1 = msg @ Wlo + blo ; out2 = LN(out1 + x)
    gemm(MSG, gt_Wlo + oDD, gt_blo + oD, Qb, Nn, Dg, Dg, 0);
    k_ln_res<<<cdiv((long)Nn * 32, T), T, 0, stream>>>(Qb, xc, gt_ln1g + oD, gt_ln1b + oD, xn, Nn);
    // FFN + LN
    gemm(xn, gt_W1 + oDH, gt_b1 + oH, Kb, Nn, Dg, HIDg, 1);
    gemm(Kb, gt_W2 + oDH, gt_b2 + oD, Vb, Nn, HIDg, Dg, 0);
    k_ln_res<<<cdiv((long)Nn * 32, T), T, 0, stream>>>(Vb, xn, gt_ln2g + oD, gt_ln2b + oD, xc, Nn);

    // SAGPool
    k_init_f32<<<cdiv((long)Nn * Dg, T), T, 0, stream>>>(MSG, Nn * Dg, 0.0f);
    k_pool_agg<<<cdiv((long)Eg * Dg, T), T, 0, stream>>>(xc, SRC, DST, MASK, MSG, Eg);
    k_pool_score<<<cdiv((long)Nn * 32, T), T, 0, stream>>>(xc, MSG, p_Wrel + oD, p_brel + i,
                                                           p_Wroot + oD, SCORE, Nn);
    k_pool_rank<<<cdiv(Nn, T), T, 0, stream>>>(SCORE, KEPT, NEWID, nper[i], kout[i], Nn);
    k_pool_copy<<<cdiv((long)Nn * Dg, T), T, 0, stream>>>(xc, SCORE, KEPT, NEWID, xn, Nn);
    k_pool_remap<<<cdiv(Eg, T), T, 0, stream>>>(SRC, DST, MASK, KEPT, NEWID, Eg);

    // readout on pooled graph (k nodes per graph), accumulated over stages
    k_readout<<<cdiv(Bg * Dg, T), T, 0, stream>>>(xn, XF, kout[i]);

    float* tmp = xc; xc = xn; xn = tmp;
  }

  // MLP head: lin -> BN -> ReLU -> lin -> BN -> ReLU -> lin
  gemm(XF, hW1, hb1, H1, Bg, 256, 128, 0);
  k_bn_relu<<<128, 64, 0, stream>>>(H1, bn1g, bn1b, 128);
  gemm(H1, hW2, hb2, H2, Bg, 128, 64, 0);
  k_bn_relu<<<64, 64, 0, stream>>>(H2, bn2g, bn2b, 64);
  k_head_final<<<1, 64, 0, stream>>>(H2, hW3, hb3, out);
}